// GraphSAGE_22574348108068
// MI455X (gfx1250) — compile-verified
//
#include <hip/hip_runtime.h>

typedef __attribute__((ext_vector_type(2))) float v2f;
typedef __attribute__((ext_vector_type(8))) float v8f;

#define N_NODES 50000
#define N_EDGES 1600000
#define DIM 64
#define EA_DIM 16
#define NTILES (N_NODES / 16)   // 3125, exact

// ---------------- degree (once) ----------------
__global__ void deg_kernel(const int* __restrict__ dst, float* __restrict__ deg) {
    int e = blockIdx.x * blockDim.x + threadIdx.x;
    if (e < N_EDGES) unsafeAtomicAdd(&deg[dst[e]], 1.0f);
}

__global__ void rdeg_kernel(const float* __restrict__ deg, float* __restrict__ rdeg) {
    int n = blockIdx.x * blockDim.x + threadIdx.x;
    if (n < N_NODES) rdeg[n] = 1.0f / fmaxf(deg[n], 1.0f);
}

// ---------------- edge phase: msg = relu(x[src] + edge_attr@emb_ea); s[dst] += msg
// 64 threads per edge (one per output dim); emb_ea (16x64) staged in LDS.
// x (12.8 MB) and s (12.8 MB) are L2-resident, so the random gather/scatter
// stays on-chip; HBM streams edge_attr + indices only.
__global__ void edge_kernel(const float* __restrict__ x,
                            const int* __restrict__ src,
                            const int* __restrict__ dst,
                            const float* __restrict__ edge_attr,
                            const float* __restrict__ emb_ea,
                            float* __restrict__ s) {
    __shared__ float W[EA_DIM * DIM];
    for (int i = threadIdx.x; i < EA_DIM * DIM; i += blockDim.x) W[i] = emb_ea[i];
    __syncthreads();

    int d = threadIdx.x & (DIM - 1);
    int e = blockIdx.x * (blockDim.x >> 6) + (threadIdx.x >> 6);
    if (e >= N_EDGES) return;

    int se = src[e];
    int de = dst[e];
    const float* ap = edge_attr + (size_t)e * EA_DIM;

    float ea = 0.0f;
#pragma unroll
    for (int k = 0; k < EA_DIM; ++k) ea = fmaf(ap[k], W[k * DIM + d], ea);

    float msg = fmaxf(x[(size_t)se * DIM + d] + ea, 0.0f);
    unsafeAtomicAdd(&s[(size_t)de * DIM + d], msg);
}

// ---------------- node phase: out = act( (s*rdeg)@Wl + bl + x@Wr )
// One wave per 16-node tile; fused K=128 GEMM via V_WMMA_F32_16X16X4_F32.
// Weights staged in LDS with K-pairs interleaved so each lane's B fragment
// {W[kr][col], W[kr+1][col]} is one contiguous ds_load_b64.
template <int RELU>
__global__ void node_kernel(const float* __restrict__ s,
                            const float* __restrict__ rdeg,
                            const float* __restrict__ x,
                            const float* __restrict__ Wl,
                            const float* __restrict__ bl,
                            const float* __restrict__ Wr,
                            float* __restrict__ out) {
    // Wp[(k>>1)*128 + col*2 + (k&1)] = Wcat[k][col], Wcat = [Wl; Wr] (128 x 64)
    __shared__ float Wp[128 * 64];
    __shared__ float B[64];
    for (int i = threadIdx.x; i < 64 * 64; i += blockDim.x) {
        int k   = i >> 6;
        int col = i & 63;
        Wp[((k >> 1) * 128) + col * 2 + (k & 1)]              = Wl[i];
        Wp[(((k + 64) >> 1) * 128) + col * 2 + ((k + 64) & 1)] = Wr[i];
    }
    if (threadIdx.x < 64) B[threadIdx.x] = bl[threadIdx.x];
    __syncthreads();

    int lane = threadIdx.x & 31;
    int tile = blockIdx.x * (blockDim.x >> 5) + (threadIdx.x >> 5);
    if (tile >= NTILES) return;   // wave-uniform: EXEC stays all-1s for WMMA

    int l15  = lane & 15;
    int kb   = (lane >> 4) << 1;          // 0 (lanes 0-15) or 2 (lanes 16-31)
    int mrow = tile * 16 + l15;           // A-matrix row handled by this lane
    float rd = rdeg[mrow];

    const float* srow = s + (size_t)mrow * DIM;
    const float* xrow = x + (size_t)mrow * DIM;

    // C/D accumulators, bias-initialized: element (vgpr v, lane l) = (M = v + 8*(l>=16), N = l&15)
    v8f acc[4];
#pragma unroll
    for (int j = 0; j < 4; ++j) {
        float b = B[j * 16 + l15];
        acc[j] = (v8f){b, b, b, b, b, b, b, b};
    }

    int colx2 = l15 * 2;   // within a K-pair row of Wp, fragment base for col tile j is j*32 + colx2

    // ---- first half: A = s * rdeg, K = 0..63 ----
#pragma unroll 8
    for (int k = 0; k < 64; k += 4) {
        int kr = k + kb;                          // even
        v2f a = *(const v2f*)&srow[kr];           // global_load_b64
        a.x *= rd;
        a.y *= rd;
        const float* wrow = &Wp[(kr >> 1) * 128];
#pragma unroll
        for (int j = 0; j < 4; ++j) {
            v2f b = *(const v2f*)&wrow[j * 32 + colx2];   // ds_load_b64, contiguous pair
            acc[j] = __builtin_amdgcn_wmma_f32_16x16x4_f32(
                false, a, false, b, (short)0, acc[j], false, false);
        }
    }

    // ---- second half: A = x, K = 64..127 ----
#pragma unroll 8
    for (int k = 0; k < 64; k += 4) {
        int kr = k + kb;                          // even
        v2f a = *(const v2f*)&xrow[kr];           // global_load_b64
        const float* wrow = &Wp[((kr + 64) >> 1) * 128];
#pragma unroll
        for (int j = 0; j < 4; ++j) {
            v2f b = *(const v2f*)&wrow[j * 32 + colx2];
            acc[j] = __builtin_amdgcn_wmma_f32_16x16x4_f32(
                false, a, false, b, (short)0, acc[j], false, false);
        }
    }

    // Store D: lane writes (M = v + 8*(lane>=16), N = j*16 + l15)
    int rbase = tile * 16 + ((lane >> 4) << 3);
#pragma unroll
    for (int v = 0; v < 8; ++v) {
#pragma unroll
        for (int j = 0; j < 4; ++j) {
            float val = acc[j][v];
            if (RELU) val = fmaxf(val, 0.0f);
            out[(size_t)(rbase + v) * DIM + j * 16 + l15] = val;
        }
    }
}

extern "C" void kernel_launch(void* const* d_in, const int* in_sizes, int n_in,
                              void* d_out, int out_size, void* d_ws, size_t ws_size,
                              hipStream_t stream) {
    const float* x     = (const float*)d_in[0];
    const int*   ei    = (const int*)d_in[1];
    const float* eattr = (const float*)d_in[2];
    const float* emb   = (const float*)d_in[3];
    const float* Wl[3] = {(const float*)d_in[4], (const float*)d_in[7], (const float*)d_in[10]};
    const float* bl[3] = {(const float*)d_in[5], (const float*)d_in[8], (const float*)d_in[11]};
    const float* Wr[3] = {(const float*)d_in[6], (const float*)d_in[9], (const float*)d_in[12]};
    const int* srcp = ei;             // edge_index row 0
    const int* dstp = ei + N_EDGES;   // edge_index row 1

    // workspace layout (all fp32): s[N*64], deg[N], rdeg[N], h0[N*64], h1[N*64] ~= 38.8 MB
    float* s    = (float*)d_ws;
    float* deg  = s + (size_t)N_NODES * DIM;
    float* rdeg = deg + N_NODES;
    float* h0   = rdeg + N_NODES;
    float* h1   = h0 + (size_t)N_NODES * DIM;

    hipMemsetAsync(deg, 0, N_NODES * sizeof(float), stream);
    deg_kernel<<<N_EDGES / 256, 256, 0, stream>>>(dstp, deg);
    rdeg_kernel<<<(N_NODES + 255) / 256, 256, 0, stream>>>(deg, rdeg);

    const float* xin[3] = {x, h0, h1};
    float*      xout[3] = {h0, h1, (float*)d_out};

    int nblocks_node = (NTILES + 7) / 8;   // 8 waves (tiles) per 256-thread block
    for (int L = 0; L < 3; ++L) {
        hipMemsetAsync(s, 0, (size_t)N_NODES * DIM * sizeof(float), stream);
        edge_kernel<<<N_EDGES / 4, 256, 0, stream>>>(xin[L], srcp, dstp, eattr, emb, s);
        if (L < 2)
            node_kernel<1><<<nblocks_node, 256, 0, stream>>>(s, rdeg, xin[L], Wl[L], bl[L], Wr[L], xout[L]);
        else
            node_kernel<0><<<nblocks_node, 256, 0, stream>>>(s, rdeg, xin[L], Wl[L], bl[L], Wr[L], xout[L]);
    }
}